// PairwiseMamba_16956530884947
// MI455X (gfx1250) — compile-verified
//
#include <hip/hip_runtime.h>
#include <math.h>
#include <stdint.h>

#define D_MODEL   16
#define D_STATE   8
#define D_CONV    2
#define D_INNER   4
#define NUM_PAIRS 36
#define BATCH     8
#define WIN       8
#define TLEN      1024
#define NSEQ      (BATCH * WIN * NUM_PAIRS)   // 2304 sequences
#define TILE      16
#define NTILE     (TLEN / TILE)               // 64 tiles of 16 timesteps

typedef float v2f __attribute__((ext_vector_type(2)));
typedef float v8f __attribute__((ext_vector_type(8)));

__device__ __forceinline__ float sigm(float x) { return 1.0f / (1.0f + __expf(-x)); }

// One wave32 per sequence. Lane mapping:
//   - WMMA A 16x4 f32:  lanes 0-15 hold M=0..15 {K0,K1}; lanes 16-31 hold M {K2,K3}
//   - WMMA B 4x16 f32:  lanes 0-15 hold N=0..15 {K0,K1}; lanes 16-31 hold N {K2,K3}
//   - WMMA C/D 16x16:   VGPR v: lanes 0-15 = row v, lanes 16-31 = row v+8; col = lane&15
//   - scan: lane = (d<<3)|s over the 4x8 state matrix
__global__ __launch_bounds__(32) void mamba_seq_kernel(
    const float* __restrict__ raw,   // (B,W,P,2,T)
    const float* __restrict__ ipw,   // (8,2)  in_proj_w
    const float* __restrict__ cw,    // (4,2)  conv_w
    const float* __restrict__ cb,    // (4,)   conv_b
    const float* __restrict__ xpw,   // (17,4) x_proj_w
    const float* __restrict__ dtw,   // (4,1)  dt_proj_w
    const float* __restrict__ dtb,   // (4,)   dt_proj_b
    const float* __restrict__ alog,  // (4,8)  A_log
    const float* __restrict__ dsk,   // (4,)   D_skip
    const float* __restrict__ opw,   // (2,4)  out_proj_w
    float* __restrict__ feat)        // (NSEQ,2) time-mean features
{
    __shared__ float s_sig[2 * TLEN];            // staged signal (ch0 | ch1)
    __shared__ float s_xz[TILE][16];             // in_proj WMMA dump: cols 0-3 = x, 4-7 = z
    __shared__ float s_xc[TILE][D_INNER];        // conv+silu(x)
    __shared__ float s_bc[TILE][16];             // x_proj WMMA dump: cols 0-7 = B, 8-15 = C
    __shared__ float s_y[TILE][D_INNER];         // gated scan output
    __shared__ float s_dtr[TILE];                // dt_rank=1 raw dt
    __shared__ float s_carry[D_INNER];           // conv carry (x at t0-1)

    const int lane = threadIdx.x;
    const int n    = blockIdx.x;
    const int col  = lane & 15;
    const bool lo  = lane < 16;

    // ---- stage the whole sequence (2*1024 f32 = 8KB) into LDS using the
    // CDNA5 async global->LDS path (ASYNCcnt), no VGPR round-trip ----------
    {
        const uint64_t gbase = (uint64_t)(uintptr_t)(raw + (size_t)n * (2 * TLEN));
        // Flat aperture for LDS carries the wave-relative LDS byte offset in
        // addr[31:0] (ISA 10.2), which is what the async op's VDST expects.
        const uint32_t lbase = (uint32_t)(uintptr_t)&s_sig[0];
        #pragma unroll
        for (int i = 0; i < (2 * TLEN * 4) / (16 * 32); ++i) {   // 16 iters
            const uint32_t byteoff = (uint32_t)(lane + 32 * i) * 16u;
            uint32_t loff  = lbase + byteoff;
            uint64_t gaddr = gbase + byteoff;
            asm volatile("global_load_async_to_lds_b128 %0, %1, off"
                         :: "v"(loff), "v"(gaddr) : "memory");
        }
    }
    if (lane < D_INNER) s_carry[lane] = 0.0f;

    // ---- per-lane parameters (overlaps with the async staging) ----
    const int dlane = lane >> 3, slane = lane & 7;   // scan (d,s)
    const int dconv = lane & 3;                      // conv channel
    const float cw0  = cw[dconv * 2 + 0];
    const float cw1  = cw[dconv * 2 + 1];
    const float cbv  = cb[dconv];
    const float dtwv = dtw[dlane];
    const float dtbv = dtb[dlane];
    const float dskv = dsk[dlane];
    const float Aneg = -__expf(alog[dlane * D_STATE + slane]);
    const float xp0 = xpw[0], xp1 = xpw[1], xp2 = xpw[2], xp3 = xpw[3]; // dt row

    // ---- wave-resident B matrices (4x16) for the three WMMAs ----
    v2f b_in, b_xp, b_out;
    b_in.x  = (lo && col < 8) ? ipw[col * 2 + 0] : 0.0f;            // B[c][d] = ipw[d][c]
    b_in.y  = (lo && col < 8) ? ipw[col * 2 + 1] : 0.0f;            // rows K=2,3 are zero
    b_xp.x  = lo ? xpw[(col + 1) * 4 + 0] : xpw[(col + 1) * 4 + 2]; // B[d][e-1] = xpw[e][d]
    b_xp.y  = lo ? xpw[(col + 1) * 4 + 1] : xpw[(col + 1) * 4 + 3];
    b_out.x = (col < 2) ? opw[col * 4 + (lo ? 0 : 2)] : 0.0f;       // B[d][c] = opw[c][d]
    b_out.y = (col < 2) ? opw[col * 4 + (lo ? 1 : 3)] : 0.0f;

    v8f acc = {0, 0, 0, 0, 0, 0, 0, 0};   // accumulates sum over t of out[t][c]
    float h = 0.0f;                        // SSM state for this lane's (d,s)

    // async staging must be complete before anyone reads s_sig
    asm volatile("s_wait_asynccnt 0x0" ::: "memory");
    __syncthreads();

    for (int tile = 0; tile < NTILE; ++tile) {
        const int t0 = tile * TILE;

        // ---- 1. in_proj: xz[t][d] = x_in[t] . ipw[d] (fp32 WMMA, K=2 padded) ----
        v2f a_in;
        a_in.x = lo ? s_sig[t0 + col] : 0.0f;            // K0 = channel 0
        a_in.y = lo ? s_sig[TLEN + t0 + col] : 0.0f;     // K1 = channel 1
        v8f d1 = {0, 0, 0, 0, 0, 0, 0, 0};
        d1 = __builtin_amdgcn_wmma_f32_16x16x4_f32(false, a_in, false, b_in,
                                                   (short)0, d1, false, false);
        __syncthreads();  // previous tile's readers of s_xz/s_bc/s_xc are done
        #pragma unroll
        for (int v = 0; v < 8; ++v) s_xz[lo ? v : v + 8][col] = d1[v];
        __syncthreads();

        // ---- 2. causal depthwise conv (k=2) + SiLU; 2 (t,d) pairs per lane ----
        {
            const int t = lane >> 2;                      // t in 0..7, d = dconv
            float xr  = s_xz[t][dconv];
            float xpv = (t > 0) ? s_xz[t - 1][dconv] : s_carry[dconv];
            float c1  = xpv * cw0 + xr * cw1 + cbv;
            s_xc[t][dconv] = c1 * sigm(c1);
            const int t2 = t + 8;                         // t in 8..15
            float xr2 = s_xz[t2][dconv];
            float c2  = s_xz[t2 - 1][dconv] * cw0 + xr2 * cw1 + cbv;
            s_xc[t2][dconv] = c2 * sigm(c2);
        }
        __syncthreads();
        // dt_raw (dt_rank == 1) and conv carry for next tile
        if (lo)
            s_dtr[col] = s_xc[col][0] * xp0 + s_xc[col][1] * xp1 +
                         s_xc[col][2] * xp2 + s_xc[col][3] * xp3;
        if (lane < D_INNER) s_carry[lane] = s_xz[15][lane];
        __syncthreads();

        // ---- 3. x_proj: B,C matrices (fp32 WMMA, K=4) ----
        v2f a_x;
        a_x.x = lo ? s_xc[col][0] : s_xc[col][2];
        a_x.y = lo ? s_xc[col][1] : s_xc[col][3];
        v8f d2 = {0, 0, 0, 0, 0, 0, 0, 0};
        d2 = __builtin_amdgcn_wmma_f32_16x16x4_f32(false, a_x, false, b_xp,
                                                   (short)0, d2, false, false);
        #pragma unroll
        for (int v = 0; v < 8; ++v) s_bc[lo ? v : v + 8][col] = d2[v];
        __syncthreads();

        // ---- 4. sequential selective scan, lane = (d,s), 16 steps ----
        #pragma unroll 4
        for (int tt = 0; tt < TILE; ++tt) {
            float dtr = s_dtr[tt];
            float xv  = s_xc[tt][dlane];
            float Bv  = s_bc[tt][slane];
            float Cv  = s_bc[tt][8 + slane];
            float zv  = s_xz[tt][4 + dlane];
            float u   = dtr * dtwv + dtbv;
            float dt  = (u > 20.0f) ? u : __logf(1.0f + __expf(u));   // softplus
            float dA  = __expf(dt * Aneg);
            h = dA * h + dt * Bv * xv;
            float c = h * Cv;                 // reduce over the 8 state lanes
            c += __shfl_xor(c, 1);
            c += __shfl_xor(c, 2);
            c += __shfl_xor(c, 4);
            if (slane == 0) {
                float y = (c + xv * dskv) * (zv * sigm(zv));  // skip + gate
                s_y[tt][dlane] = y;
            }
        }
        __syncthreads();

        // ---- 5. out_proj, accumulating the time-sum in the WMMA C operand ----
        v2f a_y;
        a_y.x = lo ? s_y[col][0] : s_y[col][2];
        a_y.y = lo ? s_y[col][1] : s_y[col][3];
        acc = __builtin_amdgcn_wmma_f32_16x16x4_f32(false, a_y, false, b_out,
                                                    (short)0, acc, false, false);
    }

    // ---- reduce acc rows (8 VGPRs = rows v / v+8) -> time-mean feature ----
    float fs = acc[0] + acc[1] + acc[2] + acc[3] + acc[4] + acc[5] + acc[6] + acc[7];
    fs += __shfl_xor(fs, 16);               // combine row halves of same column
    if (lane < 2) feat[(size_t)n * 2 + lane] = fs * (1.0f / (float)TLEN);
}

// relu(feat @ proj_w.T + proj_b), then mean over the 36 pairs. Deterministic
// (no atomics): one thread per (b*w, m) output element.
__global__ __launch_bounds__(256) void head_kernel(
    const float* __restrict__ feat, const float* __restrict__ pw,
    const float* __restrict__ pb, float* __restrict__ out)
{
    int gid = blockIdx.x * blockDim.x + threadIdx.x;
    if (gid >= BATCH * WIN * D_MODEL) return;
    int bw = gid / D_MODEL, m = gid % D_MODEL;
    float w0 = pw[m * 2 + 0], w1 = pw[m * 2 + 1], b = pb[m];
    float s = 0.0f;
    #pragma unroll 4
    for (int p = 0; p < NUM_PAIRS; ++p) {
        const float* f = feat + ((size_t)bw * NUM_PAIRS + p) * 2;
        float v = f[0] * w0 + f[1] * w1 + b;
        s += fmaxf(v, 0.0f);
    }
    out[gid] = s * (1.0f / (float)NUM_PAIRS);
}

extern "C" void kernel_launch(void* const* d_in, const int* in_sizes, int n_in,
                              void* d_out, int out_size, void* d_ws, size_t ws_size,
                              hipStream_t stream) {
    (void)in_sizes; (void)n_in; (void)out_size; (void)ws_size;
    const float* raw  = (const float*)d_in[0];
    const float* ipw  = (const float*)d_in[1];
    const float* cw   = (const float*)d_in[2];
    const float* cb   = (const float*)d_in[3];
    const float* xpw  = (const float*)d_in[4];
    const float* dtw  = (const float*)d_in[5];
    const float* dtb  = (const float*)d_in[6];
    const float* alog = (const float*)d_in[7];
    const float* dsk  = (const float*)d_in[8];
    const float* opw  = (const float*)d_in[9];
    const float* pw   = (const float*)d_in[10];
    const float* pb   = (const float*)d_in[11];
    float* feat = (float*)d_ws;              // NSEQ*2 floats of scratch

    mamba_seq_kernel<<<NSEQ, 32, 0, stream>>>(raw, ipw, cw, cb, xpw, dtw, dtb,
                                              alog, dsk, opw, feat);
    const int nout = BATCH * WIN * D_MODEL;  // 1024
    head_kernel<<<(nout + 255) / 256, 256, 0, stream>>>(feat, pw, pb, (float*)d_out);
}